// DeformationPerturbationLayer_21191368638656
// MI455X (gfx1250) — compile-verified
//
#include <hip/hip_runtime.h>
#include <math.h>
#include <stdint.h>

// DeformationPerturbationLayer — bilinear "localized_bulge" warp.
// B=8 (from in_sizes[0]), C=3, H=W=64. out[b,c,y,x] = bilinear(image[c], src(y,x;w[b])).
//
// CDNA5 path: stage the 48KB image into LDS with global_load_async_to_lds_b128
// (ASYNCcnt), s_wait_asynccnt 0, workgroup barrier, then sample from LDS.

#define HH 64
#define WW 64
#define NPIX (HH * WW)          // 4096
#define CC 3
#define THREADS 256             // 8 wave32 waves per workgroup
#define PIX_PER_BLOCK 1024      // 4 pixels per thread
#define IMG_BYTES (CC * NPIX * 4)   // 49152 bytes
#define B128_PER_THREAD (IMG_BYTES / (THREADS * 16))  // 12

typedef float __attribute__((address_space(3))) lds_float_t;

__global__ __launch_bounds__(THREADS)
void deform_bulge_bilinear_kernel(const float* __restrict__ wvec,
                                  const float* __restrict__ image,
                                  float* __restrict__ out)
{
    __shared__ float simg[CC * NPIX];   // 48 KB of the 320 KB/WGP LDS

    const unsigned t = threadIdx.x;
    const int tilesPerBatch = NPIX / PIX_PER_BLOCK;      // 4
    const int b    = (int)blockIdx.x / tilesPerBatch;
    const int tile = (int)blockIdx.x % tilesPerBatch;

    // ---- async-stage image -> LDS (gfx1250 GLOBAL_LOAD_ASYNC_TO_LDS_B128) ----
    {
        lds_float_t* lp = (lds_float_t*)simg;            // generic -> AS(3): raw LDS offset
        unsigned ldsBase = (unsigned)(uintptr_t)lp;
        unsigned byteOff = t * 16u;
        #pragma unroll
        for (int it = 0; it < B128_PER_THREAD; ++it) {   // 12 x 16B per lane
            unsigned lAddr = ldsBase + byteOff;
            asm volatile("global_load_async_to_lds_b128 %0, %1, %2"
                         :: "v"(lAddr), "v"(byteOff), "s"(image)
                         : "memory");
            byteOff += THREADS * 16u;
        }
        asm volatile("s_wait_asynccnt 0x0" ::: "memory");
    }
    __syncthreads();

    const float wb = wvec[b];

    const int n0 = tile * PIX_PER_BLOCK;
    #pragma unroll
    for (int k = 0; k < PIX_PER_BLOCK / THREADS; ++k) {  // 4 pixels/thread
        const int n = n0 + k * THREADS + (int)t;
        const int x = n & (WW - 1);
        const int y = n >> 6;

        // src = x - w*(x-c)*exp(-r^2 / (2*sigma^2)), sigma = 16 -> 2*sigma^2 = 512
        const float fxc = (float)x - 31.5f;
        const float fyc = (float)y - 31.5f;
        const float gau = expf(-(fxc * fxc + fyc * fyc) * (1.0f / 512.0f));
        const float sx  = (float)x - wb * fxc * gau;
        const float sy  = (float)y - wb * fyc * gau;

        // bilinear footprint == tent-weight sum relu(1-|s-j|)
        const float fj = floorf(sx), fi = floorf(sy);
        const int   j0 = (int)fj,   i0 = (int)fi;
        const int   j1 = j0 + 1,    i1 = i0 + 1;
        float fx = sx - fj, fy = sy - fi;
        float wx0 = 1.0f - fx, wx1 = fx;
        float wy0 = 1.0f - fy, wy1 = fy;

        const bool jv0 = (unsigned)j0 < (unsigned)WW;
        const bool jv1 = (unsigned)j1 < (unsigned)WW;
        const bool iv0 = (unsigned)i0 < (unsigned)HH;
        const bool iv1 = (unsigned)i1 < (unsigned)HH;
        const int cj0 = jv0 ? j0 : 0, cj1 = jv1 ? j1 : 0;
        const int ci0 = iv0 ? i0 : 0, ci1 = iv1 ? i1 : 0;
        wx0 = jv0 ? wx0 : 0.0f; wx1 = jv1 ? wx1 : 0.0f;
        wy0 = iv0 ? wy0 : 0.0f; wy1 = iv1 ? wy1 : 0.0f;

        const int o00 = ci0 * WW + cj0;
        const int o01 = ci0 * WW + cj1;
        const int o10 = ci1 * WW + cj0;
        const int o11 = ci1 * WW + cj1;

        #pragma unroll
        for (int c = 0; c < CC; ++c) {
            const float* ch = simg + c * NPIX;           // LDS gathers (ds_load)
            const float v00 = ch[o00];
            const float v01 = ch[o01];
            const float v10 = ch[o10];
            const float v11 = ch[o11];
            const float r = wy0 * (wx0 * v00 + wx1 * v01)
                          + wy1 * (wx0 * v10 + wx1 * v11);
            out[((size_t)b * CC + c) * NPIX + n] = r;
        }
    }
}

extern "C" void kernel_launch(void* const* d_in, const int* in_sizes, int n_in,
                              void* d_out, int out_size, void* d_ws, size_t ws_size,
                              hipStream_t stream) {
    const float* w     = (const float*)d_in[0];   // [B,1] -> B floats
    const float* image = (const float*)d_in[1];   // [3,64,64]
    float* out = (float*)d_out;                   // [B,3,64,64]
    const int B = in_sizes[0];                    // 8
    const int blocks = B * (NPIX / PIX_PER_BLOCK);
    deform_bulge_bilinear_kernel<<<dim3(blocks), dim3(THREADS), 0, stream>>>(w, image, out);
}